// LearnableWeightedScoring_53317724013184
// MI455X (gfx1250) — compile-verified
//
#include <hip/hip_runtime.h>
#include <hip/hip_bf16.h>

// LearnableWeightedScoring on MI455X (gfx1250, wave32, WMMA).
//
// d[b,h,i,j] = 32*<q_i,k_j> + (q_i^T W_h k_j + b_h)*Wd_h
//            = <qeff_i, k_j> + Wd_h*b_h,   qeff = 32*q + Wd_h*(q @ W_h).
//
// Kernel 1 (prep, VALU): qeff into workspace (16.8 MB).
// Kernel 2 (main, WMMA): per-(b,h) 1024x1024 = qeff @ k^T via
//   V_WMMA_F32_16X16X4_F32.
//   - k-tile (128x64) staged once per block into LDS via async
//     global->LDS copies (ASYNCcnt), stride-68 padding -> conflict-free
//     ds_load_b64 B-fragments (banks {4l,4l+1} ∪ {4l+2,4l+3} = all 64).
//   - Output streamed with non-temporal stores (256 MB written once,
//     never re-read; keeps qeff/k resident in the 192 MB L2).
// Store-bandwidth bound (256 MB @ 23.3 TB/s ~ 11 us); fp32 WMMA compute
// (~8.6 GFLOP) hides under that, so full reference precision is free.

#define EMBED_DIM 1024
#define NUM_HEADS 16
#define HEAD_DIM  64
#define BATCH     4
#define SEQ       1024
#define BT_STRIDE 68   // 64 + 4 pad: 16B-aligned rows, conflict-free b64 reads

typedef float v2f __attribute__((ext_vector_type(2)));
typedef float v8f __attribute__((ext_vector_type(8)));
typedef int   v4i __attribute__((ext_vector_type(4)));

#if __has_builtin(__builtin_amdgcn_global_load_async_to_lds_b128) && \
    __has_builtin(__builtin_amdgcn_s_wait_asynccnt)
#define HAS_ASYNC_LDS 1
#else
#define HAS_ASYNC_LDS 0
#endif

// Address-space-typed pointer casts for the async global->LDS builtin:
// param 0 is AS1 (global) v4i*, param 1 is AS3 (LDS) v4i*.
#define AS_GLOBAL_V4I(p) ((__attribute__((address_space(1))) v4i*)(p))
#define AS_LDS_V4I(p)    ((__attribute__((address_space(3))) v4i*)(p))

// ---------------------------------------------------------------------------
// Kernel 1: qeff[bh, s, e] = 32*q[bh,s,e] + Wd[h] * sum_d q[bh,s,d]*W[h,d,e]
// ---------------------------------------------------------------------------
__global__ __launch_bounds__(256)
void qeff_prep_kernel(const float* __restrict__ q,
                      const float* __restrict__ W,
                      const float* __restrict__ Wd,
                      float* __restrict__ qeff)
{
    const int idx = blockIdx.x * 256 + threadIdx.x;      // [0, B*H*S*D)
    const int e   = idx & (HEAD_DIM - 1);
    const int s   = (idx >> 6) & (SEQ - 1);
    const int bh  = idx >> 16;                           // b*H + h
    const int h   = bh & (NUM_HEADS - 1);

    const float* __restrict__ qrow = q + ((size_t)bh * SEQ + s) * HEAD_DIM;
    const float* __restrict__ Wh   = W + (size_t)h * HEAD_DIM * HEAD_DIM;

    float acc = 0.0f;
#pragma unroll
    for (int d = 0; d < HEAD_DIM; ++d)
        acc = fmaf(qrow[d], Wh[d * HEAD_DIM + e], acc);

    qeff[idx] = 32.0f * qrow[e] + Wd[h] * acc;           // sqrt(1024) == 32
}

// ---------------------------------------------------------------------------
// Kernel 2: out[bh] = qeff[bh] (1024x64) @ k[bh]^T (64x1024) + Wd[h]*b[h]
// Block: 256 threads (8 waves), 128x128 output tile.
// ---------------------------------------------------------------------------
__global__ __launch_bounds__(256)
void bilinear_score_kernel(const float* __restrict__ qeff,
                           const float* __restrict__ kmat,
                           const float* __restrict__ bvec,
                           const float* __restrict__ Wd,
                           float* __restrict__ out)
{
    __shared__ float Btile[128 * BT_STRIDE];             // 34 KB

    const int bh = blockIdx.y;                 // b*H + h
    const int h  = bh & (NUM_HEADS - 1);
    const int ti = blockIdx.x >> 3;            // row tile  [0,8)
    const int tj = blockIdx.x & 7;             // col tile  [0,8)

    const float* __restrict__ A = qeff + (size_t)bh * SEQ * HEAD_DIM;
    const float* __restrict__ K = kmat + (size_t)bh * SEQ * HEAD_DIM;
    float* __restrict__       O = out  + (size_t)bh * SEQ * SEQ;

    const int wave = threadIdx.x >> 5;         // [0,8)
    const int lane = threadIdx.x & 31;
    const int l    = lane & 15;                // M / N index within subtile
    const int half = lane >> 4;                // 0: K pair {0,1}, 1: {2,3}

    const int rowBase = ti * 128 + wave * 16;  // this wave's 16 rows
    const int colBase = tj * 128;

    // --- stage k-tile (rows colBase..+127, full K=64) into LDS: 2048 x 16B ---
    {
        const int t = threadIdx.x;
#pragma unroll
        for (int it = 0; it < 8; ++it) {
            const int c   = it * 256 + t;      // float4 chunk id [0,2048)
            const int row = c >> 4;            // [0,128)
            const int cir = c & 15;            // float4 within row
            const float* gsrc = K + (size_t)(colBase + row) * HEAD_DIM + cir * 4;
            float*       ldst = &Btile[row * BT_STRIDE + cir * 4];
#if HAS_ASYNC_LDS
            __builtin_amdgcn_global_load_async_to_lds_b128(
                AS_GLOBAL_V4I(gsrc), AS_LDS_V4I(ldst),
                /*imm offset=*/0, /*cpol=*/0);
#else
            *(float4*)ldst = *(const float4*)gsrc;
#endif
        }
#if HAS_ASYNC_LDS
        __builtin_amdgcn_s_wait_asynccnt(0);
#endif
        __syncthreads();
    }

    const float bias = Wd[h] * bvec[h];

    // --- preload all 16 A fragments for this wave's rows (32 VGPRs) ---
    // A frag (16x4 f32): lane l holds A[l][kbase], A[l][kbase+1],
    // kbase = 4*kk + 2*half  -> contiguous 8B global load.
    v2f afrag[16];
#pragma unroll
    for (int kk = 0; kk < 16; ++kk) {
        const int kbase = kk * 4 + half * 2;
        afrag[kk] = *(const v2f*)(A + (size_t)(rowBase + l) * HEAD_DIM + kbase);
    }

    // --- 8 column subtiles; each accumulates K=64 in 16 WMMA steps ---
    v8f acc[8];
#pragma unroll
    for (int n = 0; n < 8; ++n) {
#pragma unroll
        for (int r = 0; r < 8; ++r) acc[n][r] = bias;   // D = A*B + C chain
    }

#pragma unroll
    for (int n = 0; n < 8; ++n) {
        const int rowLds = (n * 16 + l) * BT_STRIDE;    // B column j in LDS
#pragma unroll
        for (int kk = 0; kk < 16; ++kk) {
            const int kbase = kk * 4 + half * 2;
            // B frag (4x16) of k^T: lane l holds k[j][kbase..kbase+1]
            // -> conflict-free ds_load_b64 (stride 68).
            v2f bfrag = *(const v2f*)(&Btile[rowLds + kbase]);
            acc[n] = __builtin_amdgcn_wmma_f32_16x16x4_f32(
                /*neg_a=*/false, afrag[kk],
                /*neg_b=*/false, bfrag,
                /*c_mod=*/(short)0, acc[n],
                /*reuse_a=*/false, /*reuse_b=*/false);
        }
    }

    // --- store: C/D layout: VGPR r -> row (r + 8*half), col = lane&15 ---
    // Non-temporal: output is written once, never re-read; don't churn L2.
#pragma unroll
    for (int n = 0; n < 8; ++n) {
        const int col = colBase + n * 16 + l;
#pragma unroll
        for (int r = 0; r < 8; ++r) {
            const int row = rowBase + r + 8 * half;
            __builtin_nontemporal_store(acc[n][r], &O[(size_t)row * SEQ + col]);
        }
    }
}

// ---------------------------------------------------------------------------
extern "C" void kernel_launch(void* const* d_in, const int* in_sizes, int n_in,
                              void* d_out, int out_size, void* d_ws, size_t ws_size,
                              hipStream_t stream)
{
    const float* q   = (const float*)d_in[0];   // [B,H,S,D]
    const float* k   = (const float*)d_in[1];   // [B,H,S,D]
    const float* W   = (const float*)d_in[2];   // [H,D,D]
    const float* b   = (const float*)d_in[3];   // [H]
    const float* Wd  = (const float*)d_in[4];   // [H]
    float*       out = (float*)d_out;           // [B,H,S,S]

    float* qeff = (float*)d_ws;                 // B*H*S*D floats = 16.8 MB

    const int nEl = BATCH * NUM_HEADS * SEQ * HEAD_DIM;
    qeff_prep_kernel<<<nEl / 256, 256, 0, stream>>>(q, W, Wd, qeff);

    dim3 grid(64, BATCH * NUM_HEADS);
    bilinear_score_kernel<<<grid, 256, 0, stream>>>(qeff, k, b, Wd, out);
}